// ESN_618475291258
// MI455X (gfx1250) — compile-verified
//
#include <hip/hip_runtime.h>
#include <hip/hip_bf16.h>

// ---------------------------------------------------------------------------
// Echo-state network recurrence on MI455X (gfx1250).
//   x_t = tanh(w_in * u_t + W @ x_{t-1}),  y_t = dot(x_t, g),  g = scatter(w_out, mask)
// W (4MB fp32) is striped 64 rows/block across 16 blocks, resident in LDS
// (row stride padded to 1028 floats -> conflict-free ds_load_b64).
// Mat-vec runs on V_WMMA_F32_16X16X4_F32 (exact fp32 matrix pipes).
// x is double-buffered in global memory; one grid barrier per step.
// ---------------------------------------------------------------------------

typedef __attribute__((ext_vector_type(2))) float v2f;
typedef __attribute__((ext_vector_type(8))) float v8f;

#define HDIM 1024
#define ROWS_PER_BLK 64
#define NBLK 16
#define TPB 256
#define LDSW 1028  // padded row stride (floats): lane n -> banks {4n..}, no conflicts

// Shared memory layout (bytes)
static constexpr int W_BYTES    = ROWS_PER_BLK * LDSW * 4;   // 263168
static constexpr int XS_OFF     = W_BYTES;                    // x_t staging (1024 f32)
static constexpr int PART_OFF   = XS_OFF + HDIM * 4;          // 8 waves * 16 partials
static constexpr int WIL_OFF    = PART_OFF + 128 * 4;         // w_in slice (64 f32)
static constexpr int GL_OFF     = WIL_OFF + 64 * 4;           // g slice (64 f32)
static constexpr int OUTACC_OFF = GL_OFF + 64 * 4;            // LDS output accumulator
static constexpr int SMEM_BYTES = OUTACC_OFF + 16;            // ~268.3 KB (< 320 KB/WGP)

// ---------------------------------------------------------------------------
// Init: zero d_out, zero x double-buffer, reset barrier, build g scatter.
// Re-run every launch so graph replay is deterministic.
// ---------------------------------------------------------------------------
__global__ void esn_init_kernel(const float* __restrict__ w_out,
                                const int* __restrict__ mask,
                                float* __restrict__ g,
                                float* __restrict__ xbuf,
                                unsigned* __restrict__ bar,
                                float* __restrict__ out, int out_n) {
  int i = blockIdx.x * blockDim.x + threadIdx.x;
  if (i < out_n) out[i] = 0.f;
  if (i < HDIM) {
    g[mask[i]] = w_out[i];   // mask is a permutation: no collisions
    xbuf[i] = 0.f;           // x_0 = 0
    xbuf[HDIM + i] = 0.f;
  }
  if (i == 0) *bar = 0u;
}

// ---------------------------------------------------------------------------
// Main recurrence kernel: 16 blocks x 256 threads (8 waves).
// Wave w: row tile (w&3) of this block's 64 rows, k-half (w>>2) of 1024.
// ---------------------------------------------------------------------------
__global__ __launch_bounds__(TPB, 1) void esn_step_kernel(
    const float* __restrict__ u, const float* __restrict__ w_in,
    const float* __restrict__ w_res, const float* __restrict__ g,
    float* __restrict__ xbuf, unsigned* __restrict__ bar,
    float* __restrict__ out, int T, int out_n) {
  extern __shared__ char smem[];
  float* Wl     = (float*)smem;
  float* xs     = (float*)(smem + XS_OFF);
  float* part   = (float*)(smem + PART_OFF);
  float* wil    = (float*)(smem + WIL_OFF);
  float* gl     = (float*)(smem + GL_OFF);
  float* outacc = (float*)(smem + OUTACC_OFF);

  const int tid   = threadIdx.x;
  const int lane  = tid & 31;
  const int wave  = tid >> 5;
  const int rbase = blockIdx.x * ROWS_PER_BLK;
  const int washout = T - out_n;
  const unsigned nblk = gridDim.x;

  // ---- One-time: stage this block's 64 rows of W into LDS (padded stride) ----
  for (int i = tid; i < ROWS_PER_BLK * (HDIM / 4); i += TPB) {
    int r  = i >> 8;             // row 0..63
    int kc = (i & 255) << 2;     // k chunk of 4 floats
    float4 v = *(const float4*)(w_res + (size_t)(rbase + r) * HDIM + kc);
    *(float4*)(Wl + r * LDSW + kc) = v;
  }
  if (tid < ROWS_PER_BLK) {
    wil[tid] = w_in[rbase + tid];
    gl[tid]  = g[rbase + tid];
  }
  if (tid == 0) *outacc = 0.f;
  __syncthreads();

  // WMMA operand addressing (transposed form: D row 0 = y across lanes 0..15)
  const int tile  = wave & 3;        // 16-row tile within block
  const int khalf = wave >> 2;       // k in [khalf*512, khalf*512+512)
  const int hi    = lane >> 4;       // 0: K={0,1}, 1: K={2,3}
  const int n     = lane & 15;       // output row within tile / A-matrix row
  const bool xlane = (n == 0);       // only A row 0 carries x
  const float* wp0 = Wl + (tile * 16 + n) * LDSW + khalf * 512 + (hi << 1);
  const float* xp0 = xs + khalf * 512 + (hi << 1);

  for (int t = 0; t < T; ++t) {
    // stage x_t from global double-buffer into LDS
    const float* xsrc = xbuf + (size_t)(t & 1) * HDIM;
    for (int j = tid; j < HDIM; j += TPB) xs[j] = xsrc[j];
    __syncthreads();

    // ---- y_tile += W_tile @ x over this wave's 512 k's (128 x K=4 WMMAs) ----
    v8f acc = {0.f, 0.f, 0.f, 0.f, 0.f, 0.f, 0.f, 0.f};
    const float* wp = wp0;
    const float* xp = xp0;
#pragma unroll 8
    for (int c = 0; c < 128; ++c) {
      v2f b  = *(const v2f*)wp;   // B[k,n] = W[row, k0+k]: ds_load_b64, no conflicts
      v2f xv = *(const v2f*)xp;   // safe address for all lanes; masked below
      v2f a;
      a.x = xlane ? xv.x : 0.f;   // A row 0 = x chunk (lanes 0 and 16 only)
      a.y = xlane ? xv.y : 0.f;
#if __has_builtin(__builtin_amdgcn_wmma_f32_16x16x4_f32)
      acc = __builtin_amdgcn_wmma_f32_16x16x4_f32(
          false, a, false, b, (short)0, acc, false, false);
#else
      acc[0] += a.x * b.x + a.y * b.y;  // host-pass / fallback placeholder
#endif
      wp += 4;
      xp += 4;
    }
    // D row 0 lives in acc[0] across lanes 0..15 (N = lane)
    if (lane < 16) part[wave * 16 + lane] = acc[0];
    __syncthreads();

    // ---- epilogue: one thread per owned row ----
    if (tid < ROWS_PER_BLK) {
      float y  = part[tid] + part[tid + 64];       // combine the two k-halves
      float ut = u[t];
      float xn = tanhf(wil[tid] * ut + y);
      xbuf[(size_t)((t + 1) & 1) * HDIM + rbase + tid] = xn;
      if (t >= washout) atomicAdd(outacc, xn * gl[tid]);   // ds_add_f32
    }
    __syncthreads();

    // ---- flush output partial + grid barrier (release/acquire) ----
    if (tid == 0) {
      if (t >= washout) atomicAdd(out + (t - washout), *outacc);
      *outacc = 0.f;
      __builtin_prefetch(u + t + 16, 0, 0);   // global_prefetch_b8 on u stream
      __threadfence();
      unsigned tgt = (unsigned)(t + 1) * nblk;
      __hip_atomic_fetch_add(bar, 1u, __ATOMIC_RELEASE, __HIP_MEMORY_SCOPE_AGENT);
      while (__hip_atomic_load(bar, __ATOMIC_ACQUIRE, __HIP_MEMORY_SCOPE_AGENT) < tgt) {
        __builtin_amdgcn_s_sleep(1);
      }
    }
    __syncthreads();
  }
}

// ---------------------------------------------------------------------------
extern "C" void kernel_launch(void* const* d_in, const int* in_sizes, int n_in,
                              void* d_out, int out_size, void* d_ws, size_t ws_size,
                              hipStream_t stream) {
  const float* u     = (const float*)d_in[0];
  const float* w_in  = (const float*)d_in[1];
  const float* w_res = (const float*)d_in[2];
  const float* w_out = (const float*)d_in[3];
  const int*   mask  = (const int*)d_in[4];
  float* out = (float*)d_out;

  char* ws = (char*)d_ws;
  float*    g    = (float*)ws;                         // 1024 f32
  float*    xbuf = (float*)(ws + HDIM * 4);            // 2 x 1024 f32 double-buffer
  unsigned* bar  = (unsigned*)(ws + HDIM * 4 + 2 * HDIM * 4);

  const int T = in_sizes[0];

  int initn = (out_size > HDIM ? out_size : HDIM);
  esn_init_kernel<<<(initn + TPB - 1) / TPB, TPB, 0, stream>>>(
      w_out, mask, g, xbuf, bar, out, out_size);

  (void)hipFuncSetAttribute(reinterpret_cast<const void*>(esn_step_kernel),
                            hipFuncAttributeMaxDynamicSharedMemorySize, SMEM_BYTES);

  esn_step_kernel<<<NBLK, TPB, SMEM_BYTES, stream>>>(
      u, w_in, w_res, g, xbuf, bar, out, T, out_size);
}